// weightedContrastiveLoss_73349451481503
// MI455X (gfx1250) — compile-verified
//
#include <hip/hip_runtime.h>

#define BN 4096
#define DK 1024
#define MARGIN 0.3f
#define LAMB 10.0f
#define EPSV 1e-12f
#define N_ITERS 200
#define U0 (1.0f / 4096.0f)

typedef __attribute__((ext_vector_type(2))) float v2f;
typedef __attribute__((ext_vector_type(8))) float v8f;

__device__ __forceinline__ unsigned umax_(unsigned a, unsigned b) { return a > b ? a : b; }
__device__ __forceinline__ unsigned umin_(unsigned a, unsigned b) { return a < b ? a : b; }

// map float -> order-preserving unsigned key (works for all finite floats)
__device__ __forceinline__ unsigned f2sort(float f) {
    unsigned b = __float_as_uint(f);
    return (b & 0x80000000u) ? ~b : (b | 0x80000000u);
}
__device__ __forceinline__ float sort2f(unsigned s) {
    unsigned b = (s & 0x80000000u) ? (s ^ 0x80000000u) : ~s;
    return __uint_as_float(b);
}

// ---------- init: u = 1, max/min accumulators ----------
__global__ void init_ws(float* __restrict__ u, unsigned* __restrict__ mm) {
    int t = blockIdx.x * 256 + threadIdx.x;
    if (t < BN) u[t] = 1.0f;
    if (t == 0) { mm[0] = 0u; mm[1] = 0xFFFFFFFFu; }
}

// ---------- row sum of squares ----------
__global__ void row_sq(const float* __restrict__ x, float* __restrict__ sq) {
    const int i = blockIdx.x;
    const float* row = x + (size_t)i * DK;
    float s = 0.f;
    for (int j = threadIdx.x; j < DK; j += 256) { float a = row[j]; s += a * a; }
    __shared__ float sm[256];
    sm[threadIdx.x] = s; __syncthreads();
    for (int off = 128; off > 0; off >>= 1) {
        if (threadIdx.x < off) sm[threadIdx.x] += sm[threadIdx.x + off];
        __syncthreads();
    }
    if (threadIdx.x == 0) sq[i] = sm[0];
}

// ---------- WMMA f32 GEMM + fused ground-metric epilogue ----------
// block = 256 threads (8 waves), block tile 128(M) x 64(N), wave tile 32x32 (2x2 WMMA)
__global__ void gm_gemm(const float* __restrict__ X1, const float* __restrict__ X2,
                        const long long* __restrict__ T1, const long long* __restrict__ T2,
                        const float* __restrict__ sq1, const float* __restrict__ sq2,
                        float* __restrict__ gm, unsigned* __restrict__ mm) {
    const int lane = threadIdx.x & 31;
    const int wave = threadIdx.x >> 5;
    const int wm0 = blockIdx.x * 128 + (wave & 3) * 32;
    const int wn0 = blockIdx.y * 64 + (wave >> 2) * 32;
    const int mrow  = lane & 15;          // row (A) / col (B) within 16-tile
    const int khalf = (lane >> 4) * 2;    // K sub-offset per ISA 32-bit A layout

    v8f c[2][2] = {{{0.f}, {0.f}}, {{0.f}, {0.f}}};

    for (int k = 0; k < DK; k += 4) {
        v2f a[2], b[2];
#pragma unroll
        for (int t = 0; t < 2; ++t) {
            a[t] = *(const v2f*)(X1 + (size_t)(wm0 + t * 16 + mrow) * DK + k + khalf);
            b[t] = *(const v2f*)(X2 + (size_t)(wn0 + t * 16 + mrow) * DK + k + khalf);
        }
#pragma unroll
        for (int tm = 0; tm < 2; ++tm)
#pragma unroll
            for (int tn = 0; tn < 2; ++tn)
                c[tm][tn] = __builtin_amdgcn_wmma_f32_16x16x4_f32(
                    false, a[tm], false, b[tn], (short)0, c[tm][tn], false, false);
    }

    // epilogue: gm = sim ? d2 : max(0, MARGIN - d2); track global max/min
    unsigned lmax = 0u, lmin = 0xFFFFFFFFu;
    const int colL  = lane & 15;
    const int rowHi = (lane >> 4) * 8;
#pragma unroll
    for (int tm = 0; tm < 2; ++tm) {
#pragma unroll
        for (int tn = 0; tn < 2; ++tn) {
#pragma unroll
            for (int r = 0; r < 8; ++r) {
                const int row = wm0 + tm * 16 + rowHi + r;
                const int col = wn0 + tn * 16 + colL;
                const float raw = sq1[row] - 2.0f * c[tm][tn][r] + sq2[col];
                const float g = (T1[row] == T2[col]) ? raw : fmaxf(0.0f, MARGIN - raw);
                gm[(size_t)row * BN + col] = g;
                const unsigned s = f2sort(g);
                lmax = umax_(lmax, s);
                lmin = umin_(lmin, s);
            }
        }
    }
    __shared__ unsigned smax[256], smin[256];
    smax[threadIdx.x] = lmax; smin[threadIdx.x] = lmin;
    __syncthreads();
    for (int off = 128; off > 0; off >>= 1) {
        if (threadIdx.x < off) {
            smax[threadIdx.x] = umax_(smax[threadIdx.x], smax[threadIdx.x + off]);
            smin[threadIdx.x] = umin_(smin[threadIdx.x], smin[threadIdx.x + off]);
        }
        __syncthreads();
    }
    if (threadIdx.x == 0) {
        atomicMax(&mm[0], smax[0]);
        atomicMin(&mm[1], smin[0]);
    }
}

// ---------- derive K(g) = exp(coef*(g - maxg)) parameters ----------
__global__ void scale_params(const unsigned* __restrict__ mm, float* __restrict__ params) {
    if (threadIdx.x == 0) {
        const float maxg = sort2f(mm[0]);
        const float ming = sort2f(mm[1]);
        params[0] = maxg;
        params[1] = LAMB / (maxg - ming + EPSV);  // K = exp(-LAMB*(maxg-g)/(maxg-ming+eps))
    }
}

// ---------- Sinkhorn: v = u0 / (K^T u + eps), split-K stage 1 ----------
// grid(16 colblocks, 16 rowslices); coalesced across contiguous columns
__global__ void col_partial(const float* __restrict__ gm, const float* __restrict__ params,
                            const float* __restrict__ uin, float* __restrict__ part) {
    const int col = blockIdx.x * 256 + threadIdx.x;
    const int r0  = blockIdx.y * 256;
    const float maxg = params[0], coef = params[1];
    float s = 0.f;
    for (int ii = 0; ii < 256; ++ii) {
        const int i = r0 + ii;
        s += __expf(coef * (gm[(size_t)i * BN + col] - maxg)) * uin[i];
    }
    part[blockIdx.y * BN + col] = s;
}

__global__ void col_reduce(const float* __restrict__ part, float* __restrict__ vout) {
    const int j = blockIdx.x * 256 + threadIdx.x;
    float s = 0.f;
    for (int r = 0; r < 16; ++r) s += part[r * BN + j];
    vout[j] = U0 / (s + EPSV);
}

// ---------- Sinkhorn: u = u0 / (K v + eps), block-per-row ----------
__global__ void row_matvec(const float* __restrict__ gm, const float* __restrict__ params,
                           const float* __restrict__ vin, float* __restrict__ uout) {
    const int i = blockIdx.x;
    const float maxg = params[0], coef = params[1];
    const float* row = gm + (size_t)i * BN;
    float s = 0.f;
    for (int j = threadIdx.x; j < BN; j += 256)
        s += __expf(coef * (row[j] - maxg)) * vin[j];
    __shared__ float sm[256];
    sm[threadIdx.x] = s; __syncthreads();
    for (int off = 128; off > 0; off >>= 1) {
        if (threadIdx.x < off) sm[threadIdx.x] += sm[threadIdx.x + off];
        __syncthreads();
    }
    if (threadIdx.x == 0) uout[i] = U0 / (sm[0] + EPSV);
}

// ---------- loss = sum_ij gm * u_i * K_ij * v_j ----------
__global__ void loss_row(const float* __restrict__ gm, const float* __restrict__ params,
                         const float* __restrict__ u, const float* __restrict__ v,
                         float* __restrict__ rowp) {
    const int i = blockIdx.x;
    const float maxg = params[0], coef = params[1];
    const float* row = gm + (size_t)i * BN;
    float s = 0.f;
    for (int j = threadIdx.x; j < BN; j += 256) {
        const float g = row[j];
        s += g * __expf(coef * (g - maxg)) * v[j];
    }
    __shared__ float sm[256];
    sm[threadIdx.x] = s; __syncthreads();
    for (int off = 128; off > 0; off >>= 1) {
        if (threadIdx.x < off) sm[threadIdx.x] += sm[threadIdx.x + off];
        __syncthreads();
    }
    if (threadIdx.x == 0) rowp[i] = u[i] * sm[0];
}

__global__ void loss_sum(const float* __restrict__ rowp, float* __restrict__ out) {
    float s = 0.f;
    for (int i = threadIdx.x; i < BN; i += 256) s += rowp[i];
    __shared__ float sm[256];
    sm[threadIdx.x] = s; __syncthreads();
    for (int off = 128; off > 0; off >>= 1) {
        if (threadIdx.x < off) sm[threadIdx.x] += sm[threadIdx.x + off];
        __syncthreads();
    }
    if (threadIdx.x == 0) out[0] = sm[0];
}

extern "C" void kernel_launch(void* const* d_in, const int* in_sizes, int n_in,
                              void* d_out, int out_size, void* d_ws, size_t ws_size,
                              hipStream_t stream) {
    (void)in_sizes; (void)n_in; (void)out_size; (void)ws_size;
    const float*     X1 = (const float*)d_in[0];
    const float*     X2 = (const float*)d_in[1];
    const long long* T1 = (const long long*)d_in[2];
    const long long* T2 = (const long long*)d_in[3];
    float* out = (float*)d_out;

    float* ws     = (float*)d_ws;
    float* gm     = ws;                         // 4096*4096 = 64 MB
    float* sq1    = gm + (size_t)BN * BN;       // 4096
    float* sq2    = sq1 + BN;                   // 4096
    float* u      = sq2 + BN;                   // 4096
    float* v      = u + BN;                     // 4096
    float* part   = v + BN;                     // 16*4096
    float* rowp   = part + 16 * BN;             // 4096
    float* params = rowp + BN;                  // maxg, coef
    unsigned* mm  = (unsigned*)(params + 2);    // sortable max, min

    init_ws<<<16, 256, 0, stream>>>(u, mm);
    row_sq<<<BN, 256, 0, stream>>>(X1, sq1);
    row_sq<<<BN, 256, 0, stream>>>(X2, sq2);

    dim3 gg(BN / 128, BN / 64);
    gm_gemm<<<gg, 256, 0, stream>>>(X1, X2, T1, T2, sq1, sq2, gm, mm);
    scale_params<<<1, 64, 0, stream>>>(mm, params);

    for (int it = 0; it < N_ITERS; ++it) {
        col_partial<<<dim3(16, 16), 256, 0, stream>>>(gm, params, u, part);
        col_reduce<<<16, 256, 0, stream>>>(part, v);
        row_matvec<<<BN, 256, 0, stream>>>(gm, params, v, u);
    }

    loss_row<<<BN, 256, 0, stream>>>(gm, params, u, v, rowp);
    loss_sum<<<1, 256, 0, stream>>>(rowp, out);
}